// LSTM_Stack_7988639170904
// MI455X (gfx1250) — compile-verified
//
#include <hip/hip_runtime.h>

// ---------------------------------------------------------------------------
// LSTM encoder/decoder with neural-stack read, fp32, CDNA5 (gfx1250).
// Matrix math on V_WMMA_F32_16X16X4_F32 (full-precision matrix pipe).
// ---------------------------------------------------------------------------

#define BATCH 256
#define DIM_D 128
#define DIM_S 64
#define DIM_H 512
#define NGATE 2048   // 4 * H

typedef __attribute__((ext_vector_type(2))) float v2f;
typedef __attribute__((ext_vector_type(8))) float v8f;

static __device__ __forceinline__ v8f wmma4(v2f a, v2f b, v8f c) {
  // D(16x16 f32) = A(16x4 f32) * B(4x16 f32) + C
  return __builtin_amdgcn_wmma_f32_16x16x4_f32(
      /*neg_a=*/false, a, /*neg_b=*/false, b,
      /*c_mod=*/(short)0, c, /*reuse_a=*/false, /*reuse_b=*/false);
}

static __device__ __forceinline__ float sigm(float x) {
  return 1.0f / (1.0f + __expf(-x));
}
static __device__ __forceinline__ float tanh_fast(float x) {
  return 2.0f / (1.0f + __expf(-2.0f * x)) - 1.0f;
}

// One GEMM K-phase: A row-slice (pre-offset by lane K-half) against the four
// gate blocks of W (row ptrs pre-offset by lane K-half). 4x reuse of the A
// fragment per K-slice.
static __device__ __forceinline__ void gemm_phase(
    const float* __restrict__ ap,
    const float* __restrict__ wi, const float* __restrict__ wf,
    const float* __restrict__ wg, const float* __restrict__ wo,
    int klen, v8f& ci, v8f& cf, v8f& cg, v8f& co) {
  for (int k0 = 0; k0 < klen; k0 += 4) {
    v2f a; a[0] = ap[k0]; a[1] = ap[k0 + 1];
    v2f b;
    b[0] = wi[k0]; b[1] = wi[k0 + 1]; ci = wmma4(a, b, ci);
    b[0] = wf[k0]; b[1] = wf[k0 + 1]; cf = wmma4(a, b, cf);
    b[0] = wg[k0]; b[1] = wg[k0 + 1]; cg = wmma4(a, b, cg);
    b[0] = wo[k0]; b[1] = wo[k0 + 1]; co = wmma4(a, b, co);
  }
}

// gates = [x_t | r | h] @ [Wih | Whh]^T + bih + bhh, fused LSTM cell update.
// Grid: 64 blocks x 256 threads (8 waves). Wave owns one 16x16 (batch,hidden)
// tile; all 8 waves of a block share the same hidden tile (W reuse in WGP$).
__global__ __launch_bounds__(256) void lstm_gates_cell(
    const float* __restrict__ x_t,   // [256,128] or unused
    const float* __restrict__ r,     // [256, 64]
    const float* __restrict__ h_in,  // [256,512]
    const float* __restrict__ c_in,  // [256,512]
    const float* __restrict__ Wih,   // [2048,192]
    const float* __restrict__ Whh,   // [2048,512]
    const float* __restrict__ bih,   // [2048]
    const float* __restrict__ bhh,   // [2048]
    float* __restrict__ h_out,       // [256,512]
    float* __restrict__ c_out,       // [256,512]
    int has_x) {
  const int lane = threadIdx.x & 31;
  const int wave = threadIdx.x >> 5;
  const int nt = blockIdx.x >> 1;                      // hidden tile 0..31
  const int mt = ((blockIdx.x & 1) << 3) | wave;       // batch  tile 0..15
  const int m0 = mt * 16;
  const int n0 = nt * 16;
  const int row = lane & 15;              // A: M index / B: N index
  const int kh = (lane >> 4) << 1;        // K half offset: 0 or 2

  v8f ci = {}, cf = {}, cg = {}, co = {};

  const int n = n0 + row;                 // W row within a gate block
  // Gate row pointers into Wih (cols: 0..127 -> x, 128..191 -> r)
  const float* wi_ih = Wih + (size_t)(n)*192 + kh;
  const float* wf_ih = Wih + (size_t)(n + 512) * 192 + kh;
  const float* wg_ih = Wih + (size_t)(n + 1024) * 192 + kh;
  const float* wo_ih = Wih + (size_t)(n + 1536) * 192 + kh;

  if (has_x) {  // encoder: x contribution (K = 128)
    const float* ap = x_t + (size_t)(m0 + row) * 128 + kh;
    gemm_phase(ap, wi_ih, wf_ih, wg_ih, wo_ih, 128, ci, cf, cg, co);
  }
  {  // stack-read vector contribution (K = 64), Wih cols 128..191
    const float* ap = r + (size_t)(m0 + row) * 64 + kh;
    gemm_phase(ap, wi_ih + 128, wf_ih + 128, wg_ih + 128, wo_ih + 128, 64,
               ci, cf, cg, co);
  }
  {  // recurrent contribution (K = 512)
    const float* ap = h_in + (size_t)(m0 + row) * 512 + kh;
    const float* wi = Whh + (size_t)(n)*512 + kh;
    const float* wf = Whh + (size_t)(n + 512) * 512 + kh;
    const float* wg = Whh + (size_t)(n + 1024) * 512 + kh;
    const float* wo = Whh + (size_t)(n + 1536) * 512 + kh;
    gemm_phase(ap, wi, wf, wg, wo, 512, ci, cf, cg, co);
  }

  const float bi = bih[n] + bhh[n];
  const float bf = bih[n + 512] + bhh[n + 512];
  const float bg = bih[n + 1024] + bhh[n + 1024];
  const float bo = bih[n + 1536] + bhh[n + 1536];

  // C/D layout: VGPR p, lane L -> (M = p + 8*(L>=16), N = L%16)
  const int mbase = m0 + ((lane >> 4) << 3);
#pragma unroll
  for (int p = 0; p < 8; ++p) {
    const int m = mbase + p;
    const float iv = sigm(ci[p] + bi);
    const float fv = sigm(cf[p] + bf);
    const float gv = tanh_fast(cg[p] + bg);
    const float ov = sigm(co[p] + bo);
    const float cprev = c_in[(size_t)m * 512 + n];
    const float cn = fv * cprev + iv * gv;
    const float hn = ov * tanh_fast(cn);
    c_out[(size_t)m * 512 + n] = cn;
    h_out[(size_t)m * 512 + n] = hn;
  }
}

// out = h @ ro_W^T + ro_b; stack read r = (t+1)*min(dt,1)*vt;
// decoder: log_softmax(out[:, :128]) -> d_out slice.
// Grid: 16 blocks (16 batch rows each) x 256 threads (8 waves).
__global__ __launch_bounds__(256) void readout_kernel(
    const float* __restrict__ h,    // [256,512]
    const float* __restrict__ roW,  // [194,512]
    const float* __restrict__ rob,  // [194]
    float* __restrict__ r_out,      // [256,64]
    float* __restrict__ out,        // decoder: [256,128] slice; else unused
    float tp1, int full) {
  __shared__ float lds[16][224];  // cols 0..207 used (pad vs bank conflicts)
  const int lane = threadIdx.x & 31;
  const int wave = threadIdx.x >> 5;
  const int m0 = blockIdx.x * 16;
  const int row = lane & 15;
  const int kh = (lane >> 4) << 1;
  const int ntiles = full ? 13 : 5;   // encoder needs only cols 128..207
  const int tbase = full ? 0 : 8;

  for (int ti = wave; ti < ntiles; ti += 8) {  // wave-uniform loop
    const int n0 = (tbase + ti) * 16;
    int wrow = n0 + row;
    if (wrow > 193) wrow = 193;  // clamp (cols 194..207 never consumed)
    const float* ap = h + (size_t)(m0 + row) * 512 + kh;
    const float* bp = roW + (size_t)wrow * 512 + kh;
    v8f acc = {};
    for (int k0 = 0; k0 < 512; k0 += 4) {
      v2f a; a[0] = ap[k0]; a[1] = ap[k0 + 1];
      v2f b; b[0] = bp[k0]; b[1] = bp[k0 + 1];
      acc = wmma4(a, b, acc);
    }
    const float bias = rob[wrow];
    const int mloc = (lane >> 4) << 3;
#pragma unroll
    for (int p = 0; p < 8; ++p) lds[mloc + p][n0 + row] = acc[p] + bias;
  }
  __syncthreads();

  // Stack read: r[b][s] = (t+1) * min(dt,1) * vt  (16x64 elements)
  for (int e = threadIdx.x; e < 16 * 64; e += 256) {
    const int m = e >> 6, s = e & 63;
    const float dt = lds[m][192];
    r_out[(size_t)(m0 + m) * 64 + s] =
        tp1 * fminf(dt, 1.0f) * lds[m][128 + s];
  }

  if (full) {  // log_softmax over cols 0..127, two rows per wave
#pragma unroll
    for (int rr = 0; rr < 2; ++rr) {
      const int m = wave * 2 + rr;
      const float v0 = lds[m][lane], v1 = lds[m][lane + 32];
      const float v2 = lds[m][lane + 64], v3 = lds[m][lane + 96];
      float mx = fmaxf(fmaxf(v0, v1), fmaxf(v2, v3));
      for (int off = 16; off >= 1; off >>= 1)
        mx = fmaxf(mx, __shfl_xor(mx, off, 32));
      float se = __expf(v0 - mx) + __expf(v1 - mx) +
                 __expf(v2 - mx) + __expf(v3 - mx);
      for (int off = 16; off >= 1; off >>= 1) se += __shfl_xor(se, off, 32);
      const float lse = mx + __logf(se);
      float* op = out + (size_t)(m0 + m) * 128;
      op[lane] = v0 - lse;
      op[lane + 32] = v1 - lse;
      op[lane + 64] = v2 - lse;
      op[lane + 96] = v3 - lse;
    }
  }
}

__global__ void zero_kernel(float* __restrict__ p, int n) {
  const int i = blockIdx.x * blockDim.x + threadIdx.x;
  if (i < n) p[i] = 0.0f;
}

extern "C" void kernel_launch(void* const* d_in, const int* in_sizes, int n_in,
                              void* d_out, int out_size, void* d_ws,
                              size_t ws_size, hipStream_t stream) {
  (void)in_sizes; (void)n_in; (void)out_size; (void)ws_size;
  const float* x    = (const float*)d_in[0];   // [515,256,128]
  const float* eWih = (const float*)d_in[1];   // [2048,192]
  const float* eWhh = (const float*)d_in[2];   // [2048,512]
  const float* ebih = (const float*)d_in[3];
  const float* ebhh = (const float*)d_in[4];
  const float* dWih = (const float*)d_in[5];
  const float* dWhh = (const float*)d_in[6];
  const float* dbih = (const float*)d_in[7];
  const float* dbhh = (const float*)d_in[8];
  const float* roW  = (const float*)d_in[9];   // [194,512]
  const float* rob  = (const float*)d_in[10];  // [194]
  float* out = (float*)d_out;                  // [256,256,128]
  float* ws = (float*)d_ws;

  // Workspace layout (floats): h0, c0, r (zero-init block), then h1, c1.
  const int HC = BATCH * DIM_H;    // 131072
  const int RN = BATCH * DIM_S;    // 16384
  float* h0 = ws;
  float* c0 = h0 + HC;
  float* r  = c0 + HC;
  float* h1 = r + RN;
  float* c1 = h1 + HC;

  const int zn = 2 * HC + RN;
  zero_kernel<<<(zn + 255) / 256, 256, 0, stream>>>(ws, zn);

  const int K = 256;
  float* hin = h0; float* cin = c0; float* hout = h1; float* cout = c1;

  // ---- encoder: t = 0 .. K+1 ----
  for (int t = 0; t < K + 2; ++t) {
    lstm_gates_cell<<<64, 256, 0, stream>>>(
        x + (size_t)t * BATCH * DIM_D, r, hin, cin, eWih, eWhh, ebih, ebhh,
        hout, cout, /*has_x=*/1);
    readout_kernel<<<16, 256, 0, stream>>>(hout, roW, rob, r, nullptr,
                                           (float)(t + 1), /*full=*/0);
    float* th = hin; hin = hout; hout = th;
    float* tc = cin; cin = cout; cout = tc;
  }

  // ---- decoder: t = K+2 .. 2K+1, emits log-softmax outputs ----
  for (int j = 0; j < K; ++j) {
    const int t = K + 2 + j;
    lstm_gates_cell<<<64, 256, 0, stream>>>(
        nullptr, r, hin, cin, dWih, dWhh, dbih, dbhh, hout, cout,
        /*has_x=*/0);
    readout_kernel<<<16, 256, 0, stream>>>(
        hout, roW, rob, r, out + (size_t)j * BATCH * DIM_D, (float)(t + 1),
        /*full=*/1);
    float* th = hin; hin = hout; hout = th;
    float* tc = cin; cin = cout; cout = tc;
  }
}